// PQHotShared_33938831573580
// MI455X (gfx1250) — compile-verified
//
#include <hip/hip_runtime.h>

typedef float v2f __attribute__((ext_vector_type(2)));
typedef float v8f __attribute__((ext_vector_type(8)));

#define DSUB 16
#define KCB  2048

// ---------------------------------------------------------------------------
// Kernel 1: PQ quantize+dequantize.  W:[O,I] flat as S = O*I/16 subvectors.
// One wave handles a 16-subvector tile; distances via v_wmma_f32_16x16x4_f32.
// C/D layout: lane L -> N = L&15; VGPR slot r -> M = r + 8*(L>>4).
// A/B frag:   element j (0,1), half h=L>>4 -> K = 4*kc + 2*h + j, M/N = L&15.
// ---------------------------------------------------------------------------
__global__ __launch_bounds__(256) void pq_quant_kernel(
    const float* __restrict__ W, const float* __restrict__ rs,
    const float* __restrict__ cb, float* __restrict__ Wq,
    int S, int subs_per_row, int Kcb)
{
    __shared__ float cn2[KCB];   // 8 KB: ||c||^2 per code

    const int tid = threadIdx.x;
    for (int n = tid; n < Kcb; n += 256) {
        float acc = 0.f;
        #pragma unroll
        for (int k = 0; k < DSUB; ++k) {
            const float v = cb[n * DSUB + k];
            acc += v * v;
        }
        cn2[n] = acc;
    }
    __syncthreads();

    const int wave = tid >> 5;
    const int lane = tid & 31;
    const int lo   = lane & 15;
    const int hi   = lane >> 4;

    const int tile = blockIdx.x * 8 + wave;
    const int s0   = tile * 16;
    if (s0 >= S) return;                       // wave-uniform

    const int   srow = s0 + lo;                // this lane's A-matrix row
    const float rinv = 1.0f / rs[srow / subs_per_row];

    // A fragments for K=16 (4 chunks of K=4), g = W / rs
    v2f afrag[4];
    #pragma unroll
    for (int kc = 0; kc < 4; ++kc) {
        const float2 w2 = *(const float2*)(W + srow * DSUB + kc * 4 + hi * 2);
        afrag[kc][0] = w2.x * rinv;
        afrag[kc][1] = w2.y * rinv;
    }

    float minv[8];
    int   mini[8];
    #pragma unroll
    for (int r = 0; r < 8; ++r) { minv[r] = 3.4e38f; mini[r] = 0; }

    for (int n0 = 0; n0 < Kcb; n0 += 16) {
        v8f acc = {};
        #pragma unroll
        for (int kc = 0; kc < 4; ++kc) {
            // B frag: B[k][n] = cb[n0+n][k]  (cb row-major, k consecutive)
            const float2 c2 = *(const float2*)(cb + (n0 + lo) * DSUB + kc * 4 + hi * 2);
            v2f b; b[0] = c2.x; b[1] = c2.y;
            acc = __builtin_amdgcn_wmma_f32_16x16x4_f32(
                      false, afrag[kc], false, b, (short)0, acc, false, false);
        }
        const float c2n  = cn2[n0 + lo];
        const int   nidx = n0 + lo;
        #pragma unroll
        for (int r = 0; r < 8; ++r) {
            const float dd   = c2n - 2.0f * acc[r];
            const bool  take = dd < minv[r];          // per-lane codes increase -> first-min kept
            minv[r] = take ? dd   : minv[r];
            mini[r] = take ? nidx : mini[r];
        }
    }

    // reduce argmin across the 16 lanes of each half (N dimension)
    #pragma unroll
    for (int m = 1; m <= 8; m <<= 1) {
        #pragma unroll
        for (int r = 0; r < 8; ++r) {
            const float ov = __shfl_xor(minv[r], m, 32);
            const int   oi = __shfl_xor(mini[r], m, 32);
            const bool take = (ov < minv[r]) || (ov == minv[r] && oi < mini[r]);
            minv[r] = take ? ov : minv[r];
            mini[r] = take ? oi : mini[r];
        }
    }

    // dequantize: row M = r + 8*hi, element lo (coalesced 64B per 16-lane half)
    #pragma unroll
    for (int r = 0; r < 8; ++r) {
        const int s   = s0 + r + 8 * hi;
        const float v = cb[mini[r] * DSUB + lo];
        Wq[s * DSUB + lo] = v * rs[s / subs_per_row];
    }
}

// ---------------------------------------------------------------------------
// Kernel 2: C[M,N] = A[M,64] @ B[64,N], f32 WMMA, K=64 (16 chunks of 4).
// Block = 8 waves; block tile 128(M) x 128(N); B slab staged in 32 KB LDS.
// ---------------------------------------------------------------------------
__global__ __launch_bounds__(256) void lr_gemm_kernel(
    const float* __restrict__ A, const float* __restrict__ Bm,
    float* __restrict__ C, int M, int N)
{
    __shared__ float Bs[64 * 128];               // 32 KB

    const int tid   = threadIdx.x;
    const int ncol0 = blockIdx.x * 128;

    // stage B slab [64 x 128] -> LDS (float4, coalesced)
    for (int i = tid; i < 2048; i += 256) {
        const int k = i >> 5;                    // row (K)
        const int j = (i & 31) << 2;             // col within slab
        *(float4*)(&Bs[k * 128 + j]) =
            *(const float4*)(Bm + (size_t)k * N + ncol0 + j);
    }
    __syncthreads();

    const int wave = tid >> 5;
    const int lane = tid & 31;
    const int lo   = lane & 15;
    const int hi   = lane >> 4;
    const int m0   = blockIdx.y * 128 + wave * 16;

    // full A tile 16x64 in registers (16 K-chunks), reused for 8 N-tiles
    v2f afrag[16];
    #pragma unroll
    for (int kc = 0; kc < 16; ++kc) {
        const float2 a2 = *(const float2*)(A + (size_t)(m0 + lo) * 64 + kc * 4 + hi * 2);
        afrag[kc][0] = a2.x;
        afrag[kc][1] = a2.y;
    }

    #pragma unroll
    for (int nt = 0; nt < 8; ++nt) {
        v8f acc = {};
        #pragma unroll
        for (int kc = 0; kc < 16; ++kc) {
            const int kb = kc * 4 + hi * 2;      // LDS reads: conflict-free (lo varies along row)
            v2f b;
            b[0] = Bs[(kb + 0) * 128 + nt * 16 + lo];
            b[1] = Bs[(kb + 1) * 128 + nt * 16 + lo];
            acc = __builtin_amdgcn_wmma_f32_16x16x4_f32(
                      false, afrag[kc], false, b, (short)0, acc, false, false);
        }
        float* cptr = C + (size_t)(m0 + 8 * hi) * N + ncol0 + nt * 16 + lo;
        #pragma unroll
        for (int r = 0; r < 8; ++r)
            cptr[(size_t)r * N] = acc[r];
    }
}

// ---------------------------------------------------------------------------
// Launch: inputs (setup_inputs order): U, B, rs_U, rs_B, codebook (all f32).
// ws: Uq [K_hot*r] then Bq [r*d].
// ---------------------------------------------------------------------------
extern "C" void kernel_launch(void* const* d_in, const int* in_sizes, int n_in,
                              void* d_out, int out_size, void* d_ws, size_t ws_size,
                              hipStream_t stream) {
    const float* U    = (const float*)d_in[0];
    const float* B    = (const float*)d_in[1];
    const float* rs_U = (const float*)d_in[2];
    const float* rs_B = (const float*)d_in[3];
    const float* cb   = (const float*)d_in[4];
    float* out = (float*)d_out;

    const int K_hot = in_sizes[2];             // 16384
    const int r     = in_sizes[0] / K_hot;     // 64
    const int d     = in_sizes[1] / r;         // 4096
    const int Kcb   = in_sizes[4] / DSUB;      // 2048

    float* Uq = (float*)d_ws;
    float* Bq = (float*)((char*)d_ws + (size_t)K_hot * r * sizeof(float));

    const int S_U = K_hot * r / DSUB;          // 65536 subvectors
    const int S_B = r * d / DSUB;              // 16384 subvectors

    pq_quant_kernel<<<S_U / 128, 256, 0, stream>>>(U, rs_U, cb, Uq, S_U, r / DSUB, Kcb);
    pq_quant_kernel<<<S_B / 128, 256, 0, stream>>>(B, rs_B, cb, Bq, S_B, d / DSUB, Kcb);

    dim3 grid(d / 128, K_hot / 128);
    lr_gemm_kernel<<<grid, 256, 0, stream>>>(Uq, Bq, out, K_hot, d);
}